// SparseMemory_37752762532643
// MI455X (gfx1250) — compile-verified
//
#include <hip/hip_runtime.h>

// ---------------------------------------------------------------------------
// SparseMemory: brute-force squared-L2 cdist + top-8, MI455X (gfx1250, wave32)
//   memory: [8, 100000, 128] f32   query: [8, 32, 128] f32
//   out: distances [8,32,8] f32  ++  positions [8,32,8] i32 (bit-stored)
// Bandwidth-bound (410MB stream @ 23.3TB/s ~ 17.6us). GEMM via
// V_WMMA_F32_16X16X4_F32 (full fp32 precision); distances staged in LDS per
// 1024-row chunk; fused top-K so the 100k x 32 distance matrix never hits HBM.
// B-fragments are batch-loaded into registers (32 loads in flight per wave)
// and the next pass is warmed with global_prefetch_b8.
// ---------------------------------------------------------------------------

typedef __attribute__((ext_vector_type(2))) float v2f;
typedef __attribute__((ext_vector_type(8))) float v8f;

#define NB      8        // batches
#define MROWS   100000   // memory rows per batch
#define W       128      // feature width (K dim)
#define RQ      32       // queries per batch
#define TOPK    8
#define CHUNK   1024     // memory rows per workgroup
#define WAVES   8
#define RPW     16       // rows per wave per pass
#define RPP     (WAVES * RPW)          // 128 rows per block pass
#define PASSES  (CHUNK / RPP)          // 8
#define DSTRIDE 33                     // LDS row stride (bank-conflict-free)
#define NCHUNKS ((MROWS + CHUNK - 1) / CHUNK)   // 98

__device__ __forceinline__ float f_inf() { return __builtin_inff(); }

// Insert (dist,pos) into a sorted-ascending top-8 register list.
__device__ __forceinline__ void topk_insert(float dist, int pos,
                                            float (&d)[TOPK], int (&ix)[TOPK]) {
  if (dist < d[TOPK - 1]) {
    d[TOPK - 1] = dist; ix[TOPK - 1] = pos;
#pragma unroll
    for (int j = TOPK - 1; j > 0; --j) {
      if (d[j] < d[j - 1]) {
        float td = d[j]; d[j] = d[j - 1]; d[j - 1] = td;
        int   ti = ix[j]; ix[j] = ix[j - 1]; ix[j - 1] = ti;
      }
    }
  }
}

// Each lane holds a sorted top-8 list; produce the wave-global sorted top-8
// by popping the global min 8 times (shfl_xor min-reduce with arg tracking).
// All lanes end with identical outD/outI.
__device__ __forceinline__ void wave_merge8(int lane, float (&d)[TOPK], int (&ix)[TOPK],
                                            float (&outD)[TOPK], int (&outI)[TOPK]) {
#pragma unroll
  for (int s = 0; s < TOPK; ++s) {
    float v = d[0]; int ii = ix[0]; int src = lane;
#pragma unroll
    for (int off = 16; off > 0; off >>= 1) {
      float ov = __shfl_xor(v, off, 32);
      int   oi = __shfl_xor(ii, off, 32);
      int   ol = __shfl_xor(src, off, 32);
      if (ov < v || (ov == v && ol < src)) { v = ov; ii = oi; src = ol; }
    }
    outD[s] = v; outI[s] = ii;
    if (lane == src) {  // pop my head
#pragma unroll
      for (int j = 0; j < TOPK - 1; ++j) { d[j] = d[j + 1]; ix[j] = ix[j + 1]; }
      d[TOPK - 1] = f_inf(); ix[TOPK - 1] = -1;
    }
  }
}

// ---------------------------------------------------------------------------
// Kernel 1: per-(batch,chunk) distances via WMMA + per-chunk top-8 candidates.
// grid (NCHUNKS, NB), block 256 (8 waves), dynamic LDS.
// ---------------------------------------------------------------------------
__global__ __launch_bounds__(256) void
sm_dist_topk_kernel(const float* __restrict__ memory, const float* __restrict__ query,
                    float* __restrict__ partD, int* __restrict__ partI) {
  extern __shared__ char smem_raw[];
  float* sDist = (float*)smem_raw;                 // CHUNK * DSTRIDE
  float* sQ    = sDist + CHUNK * DSTRIDE;          // RQ * W
  float* sQ2   = sQ + RQ * W;                      // RQ

  const int chunk = blockIdx.x;
  const int b     = blockIdx.y;
  const int tid   = threadIdx.x;
  const int lane  = tid & 31;
  const int wave  = tid >> 5;
  const int ln    = lane & 15;   // N / M-within-tile index
  const int h     = lane >> 4;   // K-pair selector (WMMA lane layout)

  // ---- stage queries into LDS -------------------------------------------
  const float* qb = query + (size_t)b * RQ * W;
  for (int i = tid * 4; i < RQ * W; i += 256 * 4)
    *(float4*)(&sQ[i]) = *(const float4*)(&qb[i]);
  __syncthreads();
  if (tid < RQ) {
    float s = 0.f;
#pragma unroll
    for (int k = 0; k < W; k += 4) {
      float4 v = *(const float4*)(&sQ[tid * W + k]);
      s += v.x * v.x + v.y * v.y + v.z * v.z + v.w * v.w;
    }
    sQ2[tid] = s;
  }
  __syncthreads();

  // q^2 fragments matching the C-matrix layout: M = v + 8*h
  float q2a[8], q2b[8];
#pragma unroll
  for (int v = 0; v < 8; ++v) { q2a[v] = sQ2[8 * h + v]; q2b[v] = sQ2[16 + 8 * h + v]; }

  const int chunkBase = chunk * CHUNK;
  const float* mb = memory + (size_t)b * MROWS * W;

  // ---- phase 1: WMMA distance tiles -> LDS ------------------------------
  for (int pass = 0; pass < PASSES; ++pass) {
    const int localBase = pass * RPP + wave * RPW;
    const int row       = chunkBase + localBase + ln;
    const int rowC      = row < MROWS ? row : MROWS - 1;   // clamp (masked later)
    const float* mrow   = mb + (size_t)rowC * W;

    // Batch-issue all 32 B-fragment loads (32 global_load_b64 in flight).
    // Lane layout for V_WMMA B (4x16): lane ln holds K = ks+2h, ks+2h+1.
    v2f bf[32];
#pragma unroll
    for (int j = 0; j < 32; ++j)
      bf[j] = *(const v2f*)(mrow + 4 * j + 2 * h);

    // Warm next pass's rows in cache (4 x 128B lines per 512B row covered by
    // the 32-lane x 2-prefetch pattern); no counter, purely hint.
    if (pass + 1 < PASSES) {
      const int nrow  = chunkBase + (pass + 1) * RPP + wave * RPW + ln;
      const int nrowC = nrow < MROWS ? nrow : MROWS - 1;
      const char* p = (const char*)(mb + (size_t)nrowC * W) + h * 256;
      __builtin_prefetch(p, 0, 3);
      __builtin_prefetch(p + 128, 0, 3);
    }

    // ||m||^2 partial over this lane's K-subset (combined across h below).
    float m2 = 0.f;
#pragma unroll
    for (int j = 0; j < 32; ++j)
      m2 += bf[j].x * bf[j].x + bf[j].y * bf[j].y;

    v8f acc0 = {}; v8f acc1 = {};
#pragma unroll
    for (int ks = 0; ks < W; ks += 4) {
      const int kc = ks + 2 * h;
      // A fragments (16x4, row = query): lane ln holds K = kc, kc+1
      v2f a0 = *(const v2f*)(&sQ[ln * W + kc]);
      v2f a1 = *(const v2f*)(&sQ[(16 + ln) * W + kc]);
      acc0 = __builtin_amdgcn_wmma_f32_16x16x4_f32(false, a0, false, bf[ks >> 2],
                                                   (short)0, acc0, false, false);
      acc1 = __builtin_amdgcn_wmma_f32_16x16x4_f32(false, a1, false, bf[ks >> 2],
                                                   (short)0, acc1, false, false);
    }
    m2 += __shfl_xor(m2, 16, 32);   // combine the two K-halves of this row

    float* drow = &sDist[(size_t)(localBase + ln) * DSTRIDE];
#pragma unroll
    for (int v = 0; v < 8; ++v) {
      drow[v + 8 * h]      = q2a[v] + m2 - 2.0f * acc0[v];
      drow[16 + v + 8 * h] = q2b[v] + m2 - 2.0f * acc1[v];
    }
  }
  __syncthreads();

  // ---- phase 2: per-query top-8 over this chunk -------------------------
  for (int q = wave; q < RQ; q += WAVES) {
    float d[TOPK]; int ix[TOPK];
#pragma unroll
    for (int i = 0; i < TOPK; ++i) { d[i] = f_inf(); ix[i] = -1; }
    for (int r = lane; r < CHUNK; r += 32) {
      const int grow = chunkBase + r;
      const float dist = (grow < MROWS) ? sDist[(size_t)r * DSTRIDE + q] : f_inf();
      topk_insert(dist, grow, d, ix);
    }
    float oD[TOPK]; int oI[TOPK];
    wave_merge8(lane, d, ix, oD, oI);
    if (lane == 0) {
      const size_t base = (((size_t)b * RQ + q) * NCHUNKS + chunk) * TOPK;
#pragma unroll
      for (int s = 0; s < TOPK; ++s) { partD[base + s] = oD[s]; partI[base + s] = oI[s]; }
    }
  }
}

// ---------------------------------------------------------------------------
// Kernel 2: reduce NCHUNKS*8 candidates per (b,q) to the final sorted top-8.
// grid NB*RQ blocks of 32 threads (one wave each).
// ---------------------------------------------------------------------------
__global__ __launch_bounds__(32) void
sm_reduce_kernel(const float* __restrict__ partD, const int* __restrict__ partI,
                 float* __restrict__ outD, int* __restrict__ outI) {
  const int pair = blockIdx.x;           // b*RQ + q
  const int lane = threadIdx.x;
  const float* pd = partD + (size_t)pair * NCHUNKS * TOPK;
  const int*   pi = partI + (size_t)pair * NCHUNKS * TOPK;

  float d[TOPK]; int ix[TOPK];
#pragma unroll
  for (int i = 0; i < TOPK; ++i) { d[i] = f_inf(); ix[i] = -1; }
  for (int i = lane; i < NCHUNKS * TOPK; i += 32)
    topk_insert(pd[i], pi[i], d, ix);

  float oD[TOPK]; int oI[TOPK];
  wave_merge8(lane, d, ix, oD, oI);
  if (lane == 0) {
#pragma unroll
    for (int s = 0; s < TOPK; ++s) {
      outD[(size_t)pair * TOPK + s] = oD[s];
      outI[(size_t)pair * TOPK + s] = oI[s];
    }
  }
}

// ---------------------------------------------------------------------------
extern "C" void kernel_launch(void* const* d_in, const int* in_sizes, int n_in,
                              void* d_out, int out_size, void* d_ws, size_t ws_size,
                              hipStream_t stream) {
  (void)in_sizes; (void)n_in; (void)out_size; (void)ws_size;
  const float* memory = (const float*)d_in[0];
  const float* query  = (const float*)d_in[1];

  // outputs: 2048 f32 distances, then 2048 i32 positions (bit-stored)
  float* outD = (float*)d_out;
  int*   outI = (int*)((float*)d_out + (size_t)NB * RQ * TOPK);

  // workspace: candidate (dist,idx) pairs per (b,q,chunk)
  const size_t nPart = (size_t)NB * RQ * NCHUNKS * TOPK;   // 200704
  float* partD = (float*)d_ws;
  int*   partI = (int*)(partD + nPart);

  const size_t smem = ((size_t)CHUNK * DSTRIDE + RQ * W + RQ) * sizeof(float);

  dim3 grid1(NCHUNKS, NB);
  sm_dist_topk_kernel<<<grid1, 256, smem, stream>>>(memory, query, partD, partI);
  sm_reduce_kernel<<<NB * RQ, 32, 0, stream>>>(partD, partI, outD, outI);
}